// OrderingNetV4_37125697307113
// MI455X (gfx1250) — compile-verified
//
#include <hip/hip_runtime.h>
#include <math.h>

// Problem constants (match reference)
#define BB 64
#define GG 512
#define CC 384
#define HH 768
#define NITERS 10
static constexpr float TAU = 0.1f;

typedef __attribute__((ext_vector_type(2))) float v2f;
typedef __attribute__((ext_vector_type(8))) float v8f;

__device__ __forceinline__ v8f wmma_f32(v2f a, v2f b, v8f c) {
    // D = A(16x4,f32) * B(4x16,f32) + C(16x16,f32)
    return __builtin_amdgcn_wmma_f32_16x16x4_f32(
        /*neg_a=*/false, a, /*neg_b=*/false, b,
        /*c_mod=*/(short)0, c, /*reuse_a=*/false, /*reuse_b=*/false);
}

// Async copy 16 bytes global -> LDS (GLOBAL_LOAD_ASYNC_TO_LDS_B128, ASYNCcnt).
// Generic pointers to __shared__ truncate to the LDS byte offset (flat LDS
// aperture keeps offset in addr[31:0]).
__device__ __forceinline__ void async_copy_b128(void* lds_dst, const void* gsrc) {
    uint32_t la = (uint32_t)(uintptr_t)lds_dst;
    uint64_t ga = (uint64_t)(uintptr_t)gsrc;
    asm volatile("global_load_async_to_lds_b128 %0, %1, off"
                 :: "v"(la), "v"(ga) : "memory");
}
__device__ __forceinline__ void async_wait0() {
    asm volatile("s_wait_asynccnt 0x0" ::: "memory");
}

// ---------------------------------------------------------------------------
// Kernel 1: fused MLP scorer.
// Each workgroup (256 thr = 8 waves) handles 32 consecutive rows of X=(B*G, C).
//   h  = relu(X @ W1 + b1)  staged in LDS (32 x 768)
//   LA = (h @ W2 + b2) / TAU  written to global (B*G, G)
// fp32 WMMA 16x16x4. Each wave owns BOTH 16-row m-tiles so every B fragment
// loaded from global feeds two WMMAs (halves VMEM issue per WMMA).
// ---------------------------------------------------------------------------
__global__ __launch_bounds__(256) void scorer_kernel(
    const float* __restrict__ X, const float* __restrict__ W1,
    const float* __restrict__ b1, const float* __restrict__ W2,
    const float* __restrict__ b2, float* __restrict__ LA)
{
    __shared__ float Xs[32 * CC];   // 48 KB
    __shared__ float hs[32 * HH];   // 96 KB  (total 144 KB < 320 KB WGP LDS)

    const int tid  = threadIdx.x;
    const int wave = tid >> 5;
    const int lane = tid & 31;
    const int r0   = blockIdx.x * 32;

    // ---- Phase 1: stage X tile via async global->LDS (b128 per lane) ----
    {
        const float* src = X + (size_t)r0 * CC;
        for (int i = tid; i < 32 * CC / 4; i += 256)
            async_copy_b128(&Xs[i * 4], src + (size_t)i * 4);
        async_wait0();
    }
    __syncthreads();

    const int mhalf = lane >> 4;   // 0: K lo pair, 1: K hi pair
    const int ml    = lane & 15;   // row (A) / col (B) within 16x16
    const int khoff = mhalf * 2;

    // ---- Phase 2: h = relu(X @ W1 + b1) ----
    {
        const int nbase = wave * 6 * 16;   // 8 waves x 6 N-tiles x 16 = 768
        v8f acc[12];                       // [0..5]: rows 0..15, [6..11]: rows 16..31
        #pragma unroll
        for (int t = 0; t < 12; ++t) acc[t] = (v8f){0,0,0,0,0,0,0,0};

        for (int k = 0; k < CC; k += 4) {
            const float* ap0 = &Xs[ml * CC + k + khoff];
            const float* ap1 = &Xs[(16 + ml) * CC + k + khoff];
            v2f a0; a0.x = ap0[0]; a0.y = ap0[1];
            v2f a1; a1.x = ap1[0]; a1.y = ap1[1];
            if (k + 16 < CC)
                __builtin_prefetch(&W1[(size_t)(k + 16) * HH + nbase + ml], 0, 1);
            #pragma unroll
            for (int t = 0; t < 6; ++t) {
                const int n = nbase + t * 16 + ml;
                v2f b;
                b.x = W1[(size_t)(k + khoff)     * HH + n];
                b.y = W1[(size_t)(k + khoff + 1) * HH + n];
                acc[t]     = wmma_f32(a0, b, acc[t]);
                acc[6 + t] = wmma_f32(a1, b, acc[6 + t]);
            }
        }
        #pragma unroll
        for (int t = 0; t < 6; ++t) {
            const int n = nbase + t * 16 + ml;
            const float bias = b1[n];
            #pragma unroll
            for (int i = 0; i < 8; ++i) {
                const int m = i + 8 * mhalf;
                const float v0 = acc[t][i] + bias;
                const float v1 = acc[6 + t][i] + bias;
                hs[m * HH + n]        = v0 > 0.f ? v0 : 0.f;
                hs[(16 + m) * HH + n] = v1 > 0.f ? v1 : 0.f;
            }
        }
    }
    __syncthreads();

    // ---- Phase 3: LA = (h @ W2 + b2) / TAU ----
    {
        const int nbase = wave * 4 * 16;   // 8 waves x 4 N-tiles x 16 = 512
        v8f acc[8];                        // [0..3]: rows 0..15, [4..7]: rows 16..31
        #pragma unroll
        for (int t = 0; t < 8; ++t) acc[t] = (v8f){0,0,0,0,0,0,0,0};

        for (int k = 0; k < HH; k += 4) {
            const float* ap0 = &hs[ml * HH + k + khoff];
            const float* ap1 = &hs[(16 + ml) * HH + k + khoff];
            v2f a0; a0.x = ap0[0]; a0.y = ap0[1];
            v2f a1; a1.x = ap1[0]; a1.y = ap1[1];
            if (k + 16 < HH)
                __builtin_prefetch(&W2[(size_t)(k + 16) * GG + nbase + ml], 0, 1);
            #pragma unroll
            for (int t = 0; t < 4; ++t) {
                const int n = nbase + t * 16 + ml;
                v2f b;
                b.x = W2[(size_t)(k + khoff)     * GG + n];
                b.y = W2[(size_t)(k + khoff + 1) * GG + n];
                acc[t]     = wmma_f32(a0, b, acc[t]);
                acc[4 + t] = wmma_f32(a1, b, acc[4 + t]);
            }
        }
        const float inv_tau = 1.0f / TAU;
        #pragma unroll
        for (int t = 0; t < 4; ++t) {
            const int n = nbase + t * 16 + ml;
            const float bias = b2[n];
            #pragma unroll
            for (int i = 0; i < 8; ++i) {
                const int m = i + 8 * mhalf;
                LA[(size_t)(r0 + m) * GG + n]      = (acc[t][i]     + bias) * inv_tau;
                LA[(size_t)(r0 + 16 + m) * GG + n] = (acc[4 + t][i] + bias) * inv_tau;
            }
        }
    }
}

// ---------------------------------------------------------------------------
// Kernel 2: Sinkhorn row pass: la[r,:] -= logsumexp(la[r,:])
// One block per row (contiguous 512 floats); shared-mem tree reduction.
// ---------------------------------------------------------------------------
__global__ __launch_bounds__(256) void row_lse_kernel(float* __restrict__ LA)
{
    __shared__ float red[256];
    float* row = LA + (size_t)blockIdx.x * GG;
    const int t = threadIdx.x;
    const float x0 = row[t];
    const float x1 = row[t + 256];

    red[t] = fmaxf(x0, x1);
    __syncthreads();
    for (int s = 128; s > 0; s >>= 1) {
        if (t < s) red[t] = fmaxf(red[t], red[t + s]);
        __syncthreads();
    }
    const float M = red[0];
    __syncthreads();

    red[t] = expf(x0 - M) + expf(x1 - M);
    __syncthreads();
    for (int s = 128; s > 0; s >>= 1) {
        if (t < s) red[t] += red[t + s];
        __syncthreads();
    }
    const float lse = M + logf(red[0]);
    row[t]       = x0 - lse;
    row[t + 256] = x1 - lse;
}

// ---------------------------------------------------------------------------
// Kernel 3: Sinkhorn column pass (axis -2). One thread per column; coalesced
// stride-G accesses; online (streaming) logsumexp. Fuses final exp.
// ---------------------------------------------------------------------------
__global__ __launch_bounds__(256) void col_lse_kernel(float* __restrict__ LA, int do_exp)
{
    const int b   = blockIdx.y;
    const int col = blockIdx.x * 256 + threadIdx.x;
    float* base = LA + (size_t)b * GG * GG + col;

    float m = -INFINITY, s = 0.f;
    for (int r = 0; r < GG; ++r) {
        const float x = base[(size_t)r * GG];
        const float mn = fmaxf(m, x);
        s = s * expf(m - mn) + expf(x - mn);
        m = mn;
    }
    const float lse = m + logf(s);
    if (do_exp) {
        for (int r = 0; r < GG; ++r)
            base[(size_t)r * GG] = expf(base[(size_t)r * GG] - lse);
    } else {
        for (int r = 0; r < GG; ++r)
            base[(size_t)r * GG] -= lse;
    }
}

// ---------------------------------------------------------------------------
// Kernel 4: reordered_feats[b,j,c] = sum_o F[b,o,c] * P[b,o,j]  (P^T @ F).
// Block = (b, 16-wide j tile). P slab (512x16, 32 KB) staged in LDS via async
// b128 copies; A-fragments (P^T) from LDS, B streamed from F (L2-resident).
// ---------------------------------------------------------------------------
__global__ __launch_bounds__(256) void reorder_feats_kernel(
    const float* __restrict__ F, const float* __restrict__ P,
    float* __restrict__ Out)
{
    __shared__ float Ps[GG * 16];  // 32 KB
    const int b    = blockIdx.y;
    const int m0   = blockIdx.x * 16;
    const int tid  = threadIdx.x;
    const int wave = tid >> 5;
    const int lane = tid & 31;

    const float* Pb = P + (size_t)b * GG * GG;
    for (int v = tid; v < GG * 16 / 4; v += 256) {
        const int o = v >> 2, ml4 = (v & 3) * 4;
        async_copy_b128(&Ps[o * 16 + ml4], Pb + (size_t)o * GG + m0 + ml4);
    }
    async_wait0();
    __syncthreads();

    const int mhalf = lane >> 4;
    const int ml    = lane & 15;
    const int khoff = mhalf * 2;
    const int nbase = wave * 3 * 16;  // 8 waves x 48 cols = 384

    v8f acc[3];
    #pragma unroll
    for (int t = 0; t < 3; ++t) acc[t] = (v8f){0,0,0,0,0,0,0,0};

    const float* Fb = F + (size_t)b * GG * CC;
    for (int k = 0; k < GG; k += 4) {
        v2f a;  // A = P^T : A[m][k] = Ps[k][m]
        a.x = Ps[(k + khoff)     * 16 + ml];
        a.y = Ps[(k + khoff + 1) * 16 + ml];
        #pragma unroll
        for (int t = 0; t < 3; ++t) {
            const int n = nbase + t * 16 + ml;
            v2f bb;
            bb.x = Fb[(size_t)(k + khoff)     * CC + n];
            bb.y = Fb[(size_t)(k + khoff + 1) * CC + n];
            acc[t] = wmma_f32(a, bb, acc[t]);
        }
    }
    float* Ob = Out + (size_t)b * GG * CC;
    #pragma unroll
    for (int t = 0; t < 3; ++t) {
        const int n = nbase + t * 16 + ml;
        #pragma unroll
        for (int i = 0; i < 8; ++i) {
            const int j = m0 + i + 8 * mhalf;
            Ob[(size_t)j * CC + n] = acc[t][i];
        }
    }
}

// ---------------------------------------------------------------------------
// Kernel 5: reordered_coords (N=3, too small for WMMA). One thread per new
// index j; P reads coalesced across the wave, coords broadcast (scalar).
// ---------------------------------------------------------------------------
__global__ __launch_bounds__(256) void reorder_coords_kernel(
    const float* __restrict__ Coords, const float* __restrict__ P,
    float* __restrict__ Out)
{
    const int b = blockIdx.y;
    const int j = blockIdx.x * 256 + threadIdx.x;
    const float* Pb = P + (size_t)b * GG * GG + j;
    const float* cb = Coords + (size_t)b * GG * 3;
    float a0 = 0.f, a1 = 0.f, a2 = 0.f;
    for (int o = 0; o < GG; ++o) {
        const float p = Pb[(size_t)o * GG];
        a0 += p * cb[o * 3 + 0];
        a1 += p * cb[o * 3 + 1];
        a2 += p * cb[o * 3 + 2];
    }
    float* ob = Out + (size_t)b * GG * 3 + (size_t)j * 3;
    ob[0] = a0; ob[1] = a1; ob[2] = a2;
}

// ---------------------------------------------------------------------------
extern "C" void kernel_launch(void* const* d_in, const int* in_sizes, int n_in,
                              void* d_out, int out_size, void* d_ws, size_t ws_size,
                              hipStream_t stream)
{
    (void)in_sizes; (void)n_in; (void)out_size; (void)d_ws; (void)ws_size;
    const float* coords = (const float*)d_in[0];  // (B,G,3)
    const float* feats  = (const float*)d_in[1];  // (B,G,C)
    const float* W1     = (const float*)d_in[2];  // (C,H)
    const float* b1     = (const float*)d_in[3];  // (H)
    const float* W2     = (const float*)d_in[4];  // (H,G)
    const float* b2     = (const float*)d_in[5];  // (G)

    float* out        = (float*)d_out;
    float* out_coords = out;                                    // (B,G,3)
    float* out_feats  = out + (size_t)BB * GG * 3;              // (B,G,C)
    float* P          = out_feats + (size_t)BB * GG * CC;       // (B,G,G)

    // 1) MLP scorer -> log_alpha (scores/tau) directly into P region
    scorer_kernel<<<BB * GG / 32, 256, 0, stream>>>(feats, W1, b1, W2, b2, P);

    // 2) Sinkhorn: 10 x (row normalize, col normalize); last col pass fuses exp
    for (int it = 0; it < NITERS; ++it) {
        row_lse_kernel<<<BB * GG, 256, 0, stream>>>(P);
        col_lse_kernel<<<dim3(GG / 256, BB), 256, 0, stream>>>(P, it == NITERS - 1 ? 1 : 0);
    }

    // 3) Reorder features (WMMA P^T @ F) and coords (scalar)
    reorder_feats_kernel<<<dim3(GG / 16, BB), 256, 0, stream>>>(feats, P, out_feats);
    reorder_coords_kernel<<<dim3(GG / 256, BB), 256, 0, stream>>>(coords, P, out_coords);
}